// supervised_loss_21801253994518
// MI455X (gfx1250) — compile-verified
//
#include <hip/hip_runtime.h>

// Shapes from the reference: B=1024, C=8, T=4096, all fp32.
#define C_CLASSES 8
#define T_LEN     4096
#define BLOCK     256

typedef __attribute__((ext_vector_type(2))) float v2f;
typedef __attribute__((ext_vector_type(8))) float v8f;

// Exact fp32 32-lane reduction using V_WMMA_F32_16X16X4_F32.
// A (16x4 f32): lane m<16 holds {K0=p_m, K1=0}; lane m+16 holds {K2=p_{m+16}, K3=0}.
// B = ones(4x16)  =>  D[m,n] = p_m + p_{m+16} (replicated over n).
// D layout: VGPR j = row j (lanes 0-15) / row j+8 (lanes 16-31), so summing the
// 8 D registers gives lanes 0-15: sum(rows 0..7), lanes 16-31: sum(rows 8..15).
__device__ __forceinline__ float wave_reduce32(float p) {
  v2f a;  a[0] = p;     a[1] = 0.0f;
  v2f bo; bo[0] = 1.0f; bo[1] = 1.0f;
  v8f c = {};
  c = __builtin_amdgcn_wmma_f32_16x16x4_f32(
      /*neg_a=*/false, a, /*neg_b=*/false, bo,
      /*c_mod=*/(short)0, c, /*reuse_a=*/false, /*reuse_b=*/false);
  float s = ((c[0] + c[1]) + (c[2] + c[3])) + ((c[4] + c[5]) + (c[6] + c[7]));
  return __shfl(s, 0, 32) + __shfl(s, 16, 32);  // rows 0-7 sum + rows 8-15 sum
}

// Deterministic block reduction: WMMA per wave, fixed-order LDS sum of 8 waves.
// Valid result on thread 0 only.
__device__ __forceinline__ float block_reduce(float p, float* lds) {
  const int lane = threadIdx.x & 31;
  const int wid  = threadIdx.x >> 5;
  float w = wave_reduce32(p);
  if (lane == 0) lds[wid] = w;
  __syncthreads();
  float t = 0.0f;
  if (threadIdx.x == 0) {
#pragma unroll
    for (int i = 0; i < BLOCK / 32; ++i) t += lds[i];
  }
  return t;
}

// One block per (b,c) row of decsig (4096 floats = 16 KB streamed as b128 loads).
// Selected row: accumulate (eeg - d)^2 ; other rows: accumulate d^2.
__global__ void __launch_bounds__(BLOCK)
row_sq_kernel(const float* __restrict__ eeg,
              const float* __restrict__ decsig,
              const float* __restrict__ y,
              float* __restrict__ partials) {
  __shared__ float lds[BLOCK / 32];
  const int row = blockIdx.x;   // b*C + c
  const int b   = row >> 3;
  const int c   = row & 7;

  // argmax over y[b, 0..7] (first occurrence of max, matching jnp.argmax).
  const float* yb = y + (size_t)b * C_CLASSES;
  float bestv = yb[0];
  int   besti = 0;
#pragma unroll
  for (int k = 1; k < C_CLASSES; ++k) {
    float v = yb[k];
    if (v > bestv) { bestv = v; besti = k; }
  }
  const bool sel = (c == besti);  // uniform per block -> no divergence at WMMA

  const float4* drow = reinterpret_cast<const float4*>(decsig + (size_t)row * T_LEN);
  const float4* erow = reinterpret_cast<const float4*>(eeg    + (size_t)b   * T_LEN);

  float acc = 0.0f;
  if (sel) {
#pragma unroll
    for (int i = 0; i < T_LEN / 4 / BLOCK; ++i) {
      const int j = i * BLOCK + threadIdx.x;
      const float4 d = drow[j];
      const float4 e = erow[j];
      const float x0 = e.x - d.x, x1 = e.y - d.y, x2 = e.z - d.z, x3 = e.w - d.w;
      acc = fmaf(x0, x0, fmaf(x1, x1, fmaf(x2, x2, fmaf(x3, x3, acc))));
    }
  } else {
#pragma unroll
    for (int i = 0; i < T_LEN / 4 / BLOCK; ++i) {
      const int j = i * BLOCK + threadIdx.x;
      const float4 d = drow[j];
      acc = fmaf(d.x, d.x, fmaf(d.y, d.y, fmaf(d.z, d.z, fmaf(d.w, d.w, acc))));
    }
  }

  const float t = block_reduce(acc, lds);
  if (threadIdx.x == 0) partials[blockIdx.x] = t;
}

// Single-block deterministic final reduction of the 8192 per-row partials.
__global__ void __launch_bounds__(BLOCK)
final_reduce_kernel(const float* __restrict__ partials, int n,
                    float* __restrict__ out) {
  __shared__ float lds[BLOCK / 32];
  float acc = 0.0f;
  for (int i = threadIdx.x; i < n; i += BLOCK) acc += partials[i];  // fixed order
  const float t = block_reduce(acc, lds);
  if (threadIdx.x == 0) out[0] = t;
}

extern "C" void kernel_launch(void* const* d_in, const int* in_sizes, int n_in,
                              void* d_out, int out_size, void* d_ws, size_t ws_size,
                              hipStream_t stream) {
  const float* eeg    = (const float*)d_in[0];  // [B,T]
  const float* decsig = (const float*)d_in[1];  // [B,C,T]
  const float* y      = (const float*)d_in[2];  // [B,C]
  float* out      = (float*)d_out;
  float* partials = (float*)d_ws;               // rows * 4 bytes = 32 KB

  const int B    = in_sizes[0] / T_LEN;         // 1024
  const int rows = B * C_CLASSES;               // 8192

  row_sq_kernel<<<rows, BLOCK, 0, stream>>>(eeg, decsig, y, partials);
  final_reduce_kernel<<<1, BLOCK, 0, stream>>>(partials, rows, out);
}